// SO3Convolution_37203006718461
// MI455X (gfx1250) — compile-verified
//
#include <hip/hip_runtime.h>
#include <math.h>

// ---------------------------------------------------------------------------
// SO(3) convolution pipeline for MI455X (gfx1250, wave32).
// Every dense contraction (both forward pruned-DFT stages, the per-degree
// spectral GEMM, and both inverse-DFT stages) now runs on
// v_wmma_f32_16x16x4_f32 with twiddle matrices staged pair-interleaved in LDS
// (B fragments = one ds_load_b64, A fragments = one global_load_b64).
// fp32 throughout: ~22 GFLOP vs ~0.5 GB traffic => memory bound at 23.3 TB/s,
// so f32 WMMA is free vs bf16 and keeps spectral accuracy.
// ---------------------------------------------------------------------------

typedef __attribute__((ext_vector_type(2))) float v2f;
typedef __attribute__((ext_vector_type(8))) float v8f;

#define PI_F 3.14159265358979323846f
#define WMMA4(a, b, c) \
  __builtin_amdgcn_wmma_f32_16x16x4_f32(false, (a), false, (b), (short)0, (c), false, false)

namespace so3 {

constexpr int BATCH = 8;
constexpr int F_OUTc = 32;
constexpr int NGRID = 72;
constexpr int NSPEC = 5456;              // sum (2l+1)^2, l<16
constexpr float SCALING = 1.0f / 12.0f;  // 1/sqrt(72*16*16^3/32^3)

__device__ __constant__ int c_off[17] = {
    0, 1, 10, 35, 84, 165, 286, 455, 680, 969,
    1330, 1771, 2300, 2925, 3654, 4495, 5456};

__device__ __forceinline__ void decode_s(int s, int& l, int& m, int& n) {
  int ll = 0;
  while (s >= c_off[ll + 1]) ++ll;
  int d = 2 * ll + 1;
  int r = s - c_off[ll];
  l = ll;
  m = r / d - ll;
  n = r % d - ll;
}

__device__ __forceinline__ float fipow(float x, int e) {
  float r = 1.0f;
  for (int i = 0; i < e; ++i) r *= x;
  return r;
}

__device__ __forceinline__ float lfact(int k) { return lgammaf((float)k + 1.0f); }

__device__ float wigner_d(int l, int m, int n, float beta) {
  float ch = cosf(0.5f * beta);
  float sh = sinf(0.5f * beta);
  int kmin = max(0, m - n);
  int kmax = min(l + m, l - n);
  float pre = 0.5f * (lfact(l + m) + lfact(l - m) + lfact(l + n) + lfact(l - n));
  float acc = 0.0f;
  for (int k = kmin; k <= kmax; ++k) {
    float lg = pre - lfact(k) - lfact(l + m - k) - lfact(l - n - k) - lfact(n - m + k);
    float coef = (k & 1) ? -__expf(lg) : __expf(lg);
    acc += coef * fipow(ch, 2 * l + m - n - 2 * k) * fipow(sh, n - m + 2 * k);
  }
  return acc;
}

// ---------------------------------------------------------------------------
// Workspace layout (float units)
// ---------------------------------------------------------------------------
constexpr size_t O_WAN  = 0;
constexpr size_t O_WSY  = O_WAN + (size_t)NSPEC * 64;
constexpr size_t O_DRE  = O_WSY + (size_t)NSPEC * 32;
constexpr size_t O_DIM  = O_DRE + (size_t)NSPEC * 72;
constexpr size_t O_YBRE = O_DIM + (size_t)NSPEC * 72;   // packed B operand
constexpr size_t O_YBIM = O_YBRE + (size_t)NSPEC * 512;
constexpr size_t O_XCRE = O_YBIM + (size_t)NSPEC * 512;
constexpr size_t O_XCIM = O_XCRE + (size_t)NSPEC * 128;
constexpr size_t O_R    = O_XCIM + (size_t)NSPEC * 128;
constexpr size_t SZ_T1  = 16252928;  // 128 * 1984 * 64   [bf][(j,w)][u]
constexpr size_t SZ_T2  = 7872512;   // 128 * 31 * 1984   [bf][mi][(j,w)]
constexpr size_t O_T1RE = O_R;
constexpr size_t O_T1IM = O_T1RE + SZ_T1;
constexpr size_t O_T2RE = O_T1IM + SZ_T1;
constexpr size_t O_T2IM = O_T2RE + SZ_T2;
constexpr size_t O_ZCRE = O_R;  // region R reused after the forward transform
constexpr size_t O_ZCIM = O_ZCRE + (size_t)NSPEC * 256;
constexpr size_t O_XORE = O_ZCIM + (size_t)NSPEC * 256;  // rows (j,ni,b,o) x mi
constexpr size_t O_XOIM = O_XORE + 8388608;
constexpr size_t O_X2RE = O_XOIM + 8388608;              // rows (a,j,b,o) x ni
constexpr size_t O_X2IM = O_X2RE + 8388608;
constexpr size_t WS_FLOATS = O_T2IM + SZ_T2;  // ~226 MB

// ---------------------------------------------------------------------------
// Stage 0: Wigner tables.
// ---------------------------------------------------------------------------
__global__ void k_tables(const float* __restrict__ ge, float* __restrict__ Wan,
                         float* __restrict__ Wsy, float* __restrict__ Dre,
                         float* __restrict__ Dim) {
  int tid = blockIdx.x * blockDim.x + threadIdx.x;
  if (tid >= NSPEC * 168) return;
  int s = tid / 168, col = tid % 168;
  int l, m, n;
  decode_s(s, l, m, n);
  if (col < 64) {
    int j = col;
    float beta = PI_F * (float)(2 * j + 1) / 128.0f;
    float dv = wigner_d(l, m, n, beta);
    float wq = 0.0f;
    for (int k = 0; k < 32; ++k)
      wq += sinf((float)((2 * j + 1) * (2 * k + 1)) * (PI_F / 128.0f)) / (float)(2 * k + 1);
    wq *= (2.0f / 32.0f) * sinf(PI_F * (float)(2 * j + 1) / 128.0f);
    wq *= (1.0f / 4096.0f);
    Wan[s * 64 + j] = dv * wq;
  } else if (col < 96) {
    int j = col - 64;
    float beta = PI_F * (float)(2 * j + 1) / 64.0f;
    Wsy[s * 32 + j] = wigner_d(l, m, n, beta) * (float)(2 * l + 1);
  } else {
    int g = col - 96;
    float al = ge[g * 3 + 0], be = ge[g * 3 + 1], ga = ge[g * 3 + 2];
    float dv = wigner_d(l, m, n, be);
    float ph = -((float)m * al + (float)n * ga);
    float st, ct;
    __sincosf(ph, &st, &ct);
    Dre[s * 72 + g] = dv * ct;
    Dim[s * 72 + g] = dv * st;
  }
}

// ---------------------------------------------------------------------------
// Stage 1: yc -> pair-interleaved GEMM B operand.
// ---------------------------------------------------------------------------
__global__ void k_yc(const float* __restrict__ kern, const float* __restrict__ Dre,
                     const float* __restrict__ Dim, float* __restrict__ YBre,
                     float* __restrict__ YBim) {
  int tid = blockIdx.x * blockDim.x + threadIdx.x;
  if (tid >= NSPEC * 512) return;
  int io = tid & 511;
  int s = tid >> 9;
  float re = 0.0f, im = 0.0f;
  for (int g = 0; g < NGRID; ++g) {
    float kv = kern[io * NGRID + g];
    re += kv * Dre[s * 72 + g];
    im += kv * Dim[s * 72 + g];
  }
  int l, mq, nq;
  decode_s(s, l, mq, nq);
  int d = 2 * l + 1;
  int off = c_off[l];
  int r = s - off;
  int kq = r / d, nn = r % d;
  int i = io >> 5, o = io & 31;
  int rowK = kq * 16 + i;
  int colN = nn * 32 + o;
  size_t addr = (size_t)512 * off + (size_t)(rowK >> 1) * (64 * d) + colN * 2 + (rowK & 1);
  YBre[addr] = SCALING * re;
  YBim[addr] = SCALING * im;
}

// ---------------------------------------------------------------------------
// Stage 2 (WMMA): pruned DFT v->n.  A = x rows (bf,u,j) [524288 x 64];
// B = [cos | -sin] twiddles (64 x 64, cols 31/63 pad) in LDS, pair-interleaved.
// Output stored transposed: T1[bf][(j*31+w)][u] so stage 3's K is contiguous.
// ---------------------------------------------------------------------------
__global__ __launch_bounds__(256) void k_dft1w(const float* __restrict__ x,
                                               float* __restrict__ T1re,
                                               float* __restrict__ T1im) {
  __shared__ __align__(16) float B[4096];  // (v>>1)*128 + c*2 + (v&1)
  for (int e = threadIdx.x; e < 4096; e += 256) {
    int v = e >> 6, c = e & 63;
    float val = 0.0f;
    if ((c & 31) != 31) {
      int wsig = (c & 31) - 15;
      int p = (wsig * v) & 63;
      float st, ct;
      __sincosf((PI_F / 32.0f) * (float)p, &st, &ct);
      val = (c < 32) ? ct : -st;
    }
    B[(v >> 1) * 128 + c * 2 + (v & 1)] = val;
  }
  __syncthreads();
  int wave = blockIdx.x * 8 + (threadIdx.x >> 5);
  int lane = threadIdx.x & 31;
  int lr = lane & 15, kph = lane >> 4;
  int row = wave * 16 + lr;  // (bf,u,j) row; 524288 rows, all valid
  const float* ap = x + (size_t)row * 64 + 2 * kph;
  v8f z0 = {}, z1 = {}, z2 = {}, z3 = {};
  for (int ks = 0; ks < 16; ++ks) {
    v2f a = *(const v2f*)(ap + ks * 4);
    const float* bp = &B[(ks * 2 + kph) * 128 + lr * 2];
    v2f b0 = *(const v2f*)(bp);
    v2f b1 = *(const v2f*)(bp + 32);
    v2f b2 = *(const v2f*)(bp + 64);
    v2f b3 = *(const v2f*)(bp + 96);
    z0 = WMMA4(a, b0, z0);
    z1 = WMMA4(a, b1, z1);
    z2 = WMMA4(a, b2, z2);
    z3 = WMMA4(a, b3, z3);
  }
#pragma unroll
  for (int v = 0; v < 8; ++v) {
    int r2 = wave * 16 + v + 8 * kph;
    int bf = r2 >> 12, u = (r2 >> 6) & 63, j = r2 & 63;
    size_t rb = (size_t)bf * 126976 + (size_t)j * 1984 + u;  // + w*64
    T1re[rb + (size_t)lr * 64] = z0[v];
    if (lr < 15) T1re[rb + (size_t)(16 + lr) * 64] = z1[v];
    T1im[rb + (size_t)lr * 64] = z2[v];
    if (lr < 15) T1im[rb + (size_t)(16 + lr) * 64] = z3[v];
  }
}

// ---------------------------------------------------------------------------
// Stage 3 (WMMA): pruned DFT u->m, complex.  A rows (bf,j,w) [253952 x 64],
// contiguous K (thanks to stage-2 transpose).  C/S twiddles in LDS.
//   out_re = Are*C + Aim*S ; out_im = Aim*C - Are*S
// Output in the T2 layout k_xc reads: bf*61504 + mi*1984 + (j*31+w).
// ---------------------------------------------------------------------------
__global__ __launch_bounds__(256) void k_dft2w(const float* __restrict__ T1re,
                                               const float* __restrict__ T1im,
                                               float* __restrict__ T2re,
                                               float* __restrict__ T2im) {
  __shared__ __align__(16) float C[2048];  // (u>>1)*64 + mi*2 + (u&1)
  __shared__ __align__(16) float S[2048];
  for (int e = threadIdx.x; e < 2048; e += 256) {
    int u = e >> 5, mi = e & 31;
    float cv = 0.0f, sv = 0.0f;
    if (mi != 31) {
      int p = ((mi - 15) * u) & 63;
      __sincosf((PI_F / 32.0f) * (float)p, &sv, &cv);
    }
    int addr = (u >> 1) * 64 + mi * 2 + (u & 1);
    C[addr] = cv;
    S[addr] = sv;
  }
  __syncthreads();
  int wave = blockIdx.x * 8 + (threadIdx.x >> 5);
  int lane = threadIdx.x & 31;
  int lr = lane & 15, kph = lane >> 4;
  int row = wave * 16 + lr;  // (bf,j,w) row; 253952 rows (1984 % 16 == 0)
  const float* apre = T1re + (size_t)row * 64 + 2 * kph;
  const float* apim = T1im + (size_t)row * 64 + 2 * kph;
  v8f zr0 = {}, zi0 = {}, zr1 = {}, zi1 = {};
  for (int ks = 0; ks < 16; ++ks) {
    v2f ar = *(const v2f*)(apre + ks * 4);
    v2f ai = *(const v2f*)(apim + ks * 4);
    v2f arn;
    arn.x = -ar.x;
    arn.y = -ar.y;
    const float* cp = &C[(ks * 2 + kph) * 64 + lr * 2];
    const float* sp = &S[(ks * 2 + kph) * 64 + lr * 2];
    v2f c0 = *(const v2f*)cp, c1 = *(const v2f*)(cp + 32);
    v2f s0 = *(const v2f*)sp, s1 = *(const v2f*)(sp + 32);
    zr0 = WMMA4(ar, c0, zr0);
    zr0 = WMMA4(ai, s0, zr0);
    zi0 = WMMA4(ai, c0, zi0);
    zi0 = WMMA4(arn, s0, zi0);
    zr1 = WMMA4(ar, c1, zr1);
    zr1 = WMMA4(ai, s1, zr1);
    zi1 = WMMA4(ai, c1, zi1);
    zi1 = WMMA4(arn, s1, zi1);
  }
#pragma unroll
  for (int v = 0; v < 8; ++v) {
    int r2 = wave * 16 + v + 8 * kph;
    int bf = r2 / 1984;
    int rloc = r2 - bf * 1984;
    size_t ob = (size_t)bf * 61504 + rloc;  // + mi*1984
    T2re[ob + (size_t)lr * 1984] = zr0[v];
    T2im[ob + (size_t)lr * 1984] = zi0[v];
    if (lr < 15) {
      T2re[ob + (size_t)(16 + lr) * 1984] = zr1[v];
      T2im[ob + (size_t)(16 + lr) * 1984] = zi1[v];
    }
  }
}

// ---------------------------------------------------------------------------
// Stage 4: beta contraction. xc[s,b,f] = sum_j T2[bf, m_s+15, j, n_s+15]*Wan[s,j]
// ---------------------------------------------------------------------------
__global__ void k_xc(const float* __restrict__ T2re, const float* __restrict__ T2im,
                     const float* __restrict__ Wan, float* __restrict__ xcre,
                     float* __restrict__ xcim) {
  int tid = blockIdx.x * blockDim.x + threadIdx.x;
  if (tid >= NSPEC * 128) return;
  int f = tid & 15;
  int b = (tid >> 4) & 7;
  int s = tid >> 7;
  int l, m, n;
  decode_s(s, l, m, n);
  int bf = b * 16 + f;
  int base = bf * 61504 + (m + 15) * 1984 + (n + 15);  // + j*31
  float re = 0.0f, im = 0.0f;
  for (int j = 0; j < 64; ++j) {
    float wv = Wan[s * 64 + j];
    re += T2re[base + j * 31] * wv;
    im += T2im[base + j * 31] * wv;
  }
  xcre[tid] = re;
  xcim[tid] = im;
}

// ---------------------------------------------------------------------------
// Stage 5 (WMMA): per-degree complex GEMM, unchanged from round 2.
// ---------------------------------------------------------------------------
__global__ __launch_bounds__(256) void k_gemm(
    const float* __restrict__ Xre, const float* __restrict__ Xim,
    const float* __restrict__ YBre, const float* __restrict__ YBim,
    float* __restrict__ Zre, float* __restrict__ Zim) {
  int t = blockIdx.x * 8 + (threadIdx.x >> 5);
  int lane = threadIdx.x & 31;

  int d = 1, off = 0, mt = 0, nt = 0;
  {
    int rem = t;
    for (int ll = 0; ll < 16; ++ll) {
      int dd = 2 * ll + 1;
      int cnt = ((dd + 1) >> 1) * (2 * dd);
      if (rem < cnt) {
        d = dd; off = c_off[ll];
        mt = rem / (2 * dd);
        nt = rem % (2 * dd);
        break;
      }
      rem -= cnt;
    }
  }
  int Mrows = 8 * d;
  int lr = lane & 15;
  int kph = lane >> 4;
  int row = mt * 16 + lr;
  int mA = row >> 3, bA = row & 7;
  int colN = nt * 16 + lr;

  const float* Are = Xre + (size_t)(off + mA * d) * 128 + bA * 16;
  const float* Aim = Xim + (size_t)(off + mA * d) * 128 + bA * 16;
  const float* Bre = YBre + (size_t)512 * off + colN * 2;
  const float* Bim = YBim + (size_t)512 * off + colN * 2;
  const int bstride = 64 * d;
  const int aph = 2 * kph;

  v8f zr = {};
  v8f zi = {};
  for (int k = 0; k < d; ++k) {
    __builtin_prefetch(Are + (k + 1) * 128, 0, 1);
    __builtin_prefetch(Bre + (k + 1) * 8 * bstride, 0, 1);
#pragma unroll
    for (int ki = 0; ki < 4; ++ki) {
      int ao = k * 128 + ki * 4 + aph;
      int bo = (k * 8 + ki * 2 + kph) * bstride;
      v2f ar = *(const v2f*)(Are + ao);
      v2f ai = *(const v2f*)(Aim + ao);
      v2f br = *(const v2f*)(Bre + bo);
      v2f bi = *(const v2f*)(Bim + bo);
      v2f ain;
      ain.x = -ai.x;
      ain.y = -ai.y;
      zr = WMMA4(ar, br, zr);
      zr = WMMA4(ain, bi, zr);
      zi = WMMA4(ar, bi, zi);
      zi = WMMA4(ai, br, zi);
    }
  }

  int nB = colN >> 5, oB = colN & 31;
#pragma unroll
  for (int v = 0; v < 8; ++v) {
    int r2 = mt * 16 + v + 8 * kph;
    if (r2 < Mrows) {
      int m2 = r2 >> 3, b2 = r2 & 7;
      int za = ((off + m2 * d + nB) * BATCH + b2) * F_OUTc + oB;
      Zre[za] = zr[v];
      Zim[za] = zi[v];
    }
  }
}

// ---------------------------------------------------------------------------
// Stage 6: synthesis scatter into Xo rows (j,ni,b,o) x 32 contiguous mi cols.
// ---------------------------------------------------------------------------
__global__ void k_scatter(const float* __restrict__ Zre, const float* __restrict__ Zim,
                          const float* __restrict__ Wsy, float* __restrict__ Xore,
                          float* __restrict__ Xoim) {
  int tid = blockIdx.x * blockDim.x + threadIdx.x;
  if (tid >= 31 * 32 * 31 * 8 * 32) return;
  int o = tid & 31;
  int r = tid >> 5;
  int b = r & 7; r >>= 3;
  int nii = r % 31; r /= 31;
  int j = r & 31; r >>= 5;
  int mii = r;  // 0..30
  int m = mii - 15, n = nii - 15;
  int am = m < 0 ? -m : m;
  int an = n < 0 ? -n : n;
  int lmin = am > an ? am : an;
  float re = 0.0f, im = 0.0f;
  for (int l = lmin; l < 16; ++l) {
    int dd = 2 * l + 1;
    int s = c_off[l] + (m + l) * dd + (n + l);
    float wz = Wsy[s * 32 + j];
    int za = (s * 8 + b) * 32 + o;
    re += Zre[za] * wz;
    im += Zim[za] * wz;
  }
  int mi = (m + 32) & 31;
  int ni = (n + 32) & 31;
  int xa = (((j * 32 + ni) * 8 + b) * 32 + o) * 32 + mi;
  Xore[xa] = re;
  Xoim[xa] = im;
}

// ---------------------------------------------------------------------------
// Stage 7 (WMMA): inverse DFT mi -> a.  A = Xo rows (j,ni,b,o) [262144 x 32];
// cols a (2 n-tiles); out to X2 rows (a,j,b,o) x ni contiguous.
//   out_re = Xr*C - Xi*S ; out_im = Xr*S + Xi*C
// ---------------------------------------------------------------------------
__global__ __launch_bounds__(256) void k_idftAw(const float* __restrict__ Xore,
                                                const float* __restrict__ Xoim,
                                                float* __restrict__ X2re,
                                                float* __restrict__ X2im) {
  __shared__ __align__(16) float C[1024];  // (mi>>1)*64 + a*2 + (mi&1)
  __shared__ __align__(16) float S[1024];
  for (int e = threadIdx.x; e < 1024; e += 256) {
    int mi = e >> 5, a = e & 31;
    float sv, cv;
    __sincosf((PI_F / 16.0f) * (float)((a * mi) & 31), &sv, &cv);
    int addr = (mi >> 1) * 64 + a * 2 + (mi & 1);
    C[addr] = cv;
    S[addr] = sv;
  }
  __syncthreads();
  int wave = blockIdx.x * 8 + (threadIdx.x >> 5);
  int lane = threadIdx.x & 31;
  int lr = lane & 15, kph = lane >> 4;
  int row = wave * 16 + lr;  // (j,ni,b,o)
  const float* apre = Xore + (size_t)row * 32 + 2 * kph;
  const float* apim = Xoim + (size_t)row * 32 + 2 * kph;
  v8f zr0 = {}, zi0 = {}, zr1 = {}, zi1 = {};
  for (int ks = 0; ks < 8; ++ks) {
    v2f xr = *(const v2f*)(apre + ks * 4);
    v2f xi = *(const v2f*)(apim + ks * 4);
    v2f xin;
    xin.x = -xi.x;
    xin.y = -xi.y;
    const float* cp = &C[(ks * 2 + kph) * 64 + lr * 2];
    const float* sp = &S[(ks * 2 + kph) * 64 + lr * 2];
    v2f c0 = *(const v2f*)cp, c1 = *(const v2f*)(cp + 32);
    v2f s0 = *(const v2f*)sp, s1 = *(const v2f*)(sp + 32);
    zr0 = WMMA4(xr, c0, zr0);
    zr0 = WMMA4(xin, s0, zr0);
    zi0 = WMMA4(xr, s0, zi0);
    zi0 = WMMA4(xi, c0, zi0);
    zr1 = WMMA4(xr, c1, zr1);
    zr1 = WMMA4(xin, s1, zr1);
    zi1 = WMMA4(xr, s1, zi1);
    zi1 = WMMA4(xi, c1, zi1);
  }
#pragma unroll
  for (int v = 0; v < 8; ++v) {
    int r2 = wave * 16 + v + 8 * kph;
    int j = r2 >> 13, ni = (r2 >> 8) & 31, bo = r2 & 255;
    size_t ob = (size_t)j * 8192 + (size_t)bo * 32 + ni;  // + a*262144
    X2re[(size_t)lr * 262144 + ob] = zr0[v];
    X2im[(size_t)lr * 262144 + ob] = zi0[v];
    X2re[(size_t)(16 + lr) * 262144 + ob] = zr1[v];
    X2im[(size_t)(16 + lr) * 262144 + ob] = zi1[v];
  }
}

// ---------------------------------------------------------------------------
// Stage 8 (WMMA): inverse DFT ni -> c, real part only, + bias, transpose to
// (b,o,a,j,c).  A = X2 rows (a,j,b,o) [262144 x 32].
//   out = Xr*C - Xi*S
// ---------------------------------------------------------------------------
__global__ __launch_bounds__(256) void k_idftBw(const float* __restrict__ X2re,
                                                const float* __restrict__ X2im,
                                                const float* __restrict__ bias,
                                                float* __restrict__ out) {
  __shared__ __align__(16) float C[1024];  // (ni>>1)*64 + c*2 + (ni&1)
  __shared__ __align__(16) float S[1024];
  for (int e = threadIdx.x; e < 1024; e += 256) {
    int ni = e >> 5, c = e & 31;
    float sv, cv;
    __sincosf((PI_F / 16.0f) * (float)((c * ni) & 31), &sv, &cv);
    int addr = (ni >> 1) * 64 + c * 2 + (ni & 1);
    C[addr] = cv;
    S[addr] = sv;
  }
  __syncthreads();
  int wave = blockIdx.x * 8 + (threadIdx.x >> 5);
  int lane = threadIdx.x & 31;
  int lr = lane & 15, kph = lane >> 4;
  int row = wave * 16 + lr;  // (a,j,b,o)
  const float* apre = X2re + (size_t)row * 32 + 2 * kph;
  const float* apim = X2im + (size_t)row * 32 + 2 * kph;
  v8f z0 = {}, z1 = {};
  for (int ks = 0; ks < 8; ++ks) {
    v2f xr = *(const v2f*)(apre + ks * 4);
    v2f xi = *(const v2f*)(apim + ks * 4);
    v2f xin;
    xin.x = -xi.x;
    xin.y = -xi.y;
    const float* cp = &C[(ks * 2 + kph) * 64 + lr * 2];
    const float* sp = &S[(ks * 2 + kph) * 64 + lr * 2];
    v2f c0 = *(const v2f*)cp, c1 = *(const v2f*)(cp + 32);
    v2f s0 = *(const v2f*)sp, s1 = *(const v2f*)(sp + 32);
    z0 = WMMA4(xr, c0, z0);
    z0 = WMMA4(xin, s0, z0);
    z1 = WMMA4(xr, c1, z1);
    z1 = WMMA4(xin, s1, z1);
  }
#pragma unroll
  for (int v = 0; v < 8; ++v) {
    int r2 = wave * 16 + v + 8 * kph;
    int a = r2 >> 13, j = (r2 >> 8) & 31, b = (r2 >> 5) & 7, o = r2 & 31;
    size_t ob = (size_t)b * 1048576 + (size_t)o * 32768 + (size_t)a * 1024 + j * 32;
    float bv = bias[o];
    out[ob + lr] = z0[v] + bv;
    out[ob + 16 + lr] = z1[v] + bv;
  }
}

}  // namespace so3

// ---------------------------------------------------------------------------
extern "C" void kernel_launch(void* const* d_in, const int* in_sizes, int n_in,
                              void* d_out, int out_size, void* d_ws, size_t ws_size,
                              hipStream_t stream) {
  using namespace so3;
  (void)in_sizes; (void)n_in; (void)out_size;
  const float* x = (const float*)d_in[0];
  const float* kern = (const float*)d_in[1];
  const float* bias = (const float*)d_in[2];
  const float* ge = (const float*)d_in[3];
  float* ws = (float*)d_ws;
  if (ws_size < WS_FLOATS * sizeof(float)) return;

  dim3 blk(256);
  k_tables<<<(NSPEC * 168 + 255) / 256, blk, 0, stream>>>(
      ge, ws + O_WAN, ws + O_WSY, ws + O_DRE, ws + O_DIM);
  k_yc<<<(NSPEC * 512 + 255) / 256, blk, 0, stream>>>(
      kern, ws + O_DRE, ws + O_DIM, ws + O_YBRE, ws + O_YBIM);
  k_dft1w<<<4096, blk, 0, stream>>>(x, ws + O_T1RE, ws + O_T1IM);
  k_dft2w<<<1984, blk, 0, stream>>>(
      ws + O_T1RE, ws + O_T1IM, ws + O_T2RE, ws + O_T2IM);
  k_xc<<<(NSPEC * 128 + 255) / 256, blk, 0, stream>>>(
      ws + O_T2RE, ws + O_T2IM, ws + O_WAN, ws + O_XCRE, ws + O_XCIM);
  k_gemm<<<714, blk, 0, stream>>>(
      ws + O_XCRE, ws + O_XCIM, ws + O_YBRE, ws + O_YBIM, ws + O_ZCRE, ws + O_ZCIM);
  hipMemsetAsync(ws + O_XORE, 0, (size_t)2 * 8388608 * sizeof(float), stream);
  k_scatter<<<(31 * 32 * 31 * 8 * 32 + 255) / 256, blk, 0, stream>>>(
      ws + O_ZCRE, ws + O_ZCIM, ws + O_WSY, ws + O_XORE, ws + O_XOIM);
  k_idftAw<<<2048, blk, 0, stream>>>(
      ws + O_XORE, ws + O_XOIM, ws + O_X2RE, ws + O_X2IM);
  k_idftBw<<<2048, blk, 0, stream>>>(
      ws + O_X2RE, ws + O_X2IM, bias, (float*)d_out);
}